// StableLinearNodeOperator_50173807952916
// MI455X (gfx1250) — compile-verified
//
#include <hip/hip_runtime.h>

// CDNA5 / gfx1250, wave32. Fused per-atom  out = Wt*Y*cw + b (x) s  using
// V_WMMA_F32_16X16X4_F32. Memory-bound: 282 MB total HBM traffic -> ~12us floor.
// Boundary K-chunks use clamped-address loads + cndmask zeroing (no exec branches).

typedef __attribute__((ext_vector_type(2))) float v2f;
typedef __attribute__((ext_vector_type(8))) float v8f;

#define WAVES_PER_BLOCK 8
#define LDS_PAD 18  // dwords per row: 72B stride -> 8B aligned pairs, conflict-free

template <int D>
__global__ __launch_bounds__(256) void node_linear_wmma(
    const float* __restrict__ x,   // full coefficient array [T,16]
    const float* __restrict__ W,   // [D,D] row-major (W[k*D+e])
    const float* __restrict__ b,   // [D]
    const float* __restrict__ cw,  // [16,16] row-major
    float* __restrict__ out,       // full output [T,16]
    int n_atoms, long long base)   // base element offset of this type's block
{
  constexpr int MT  = (D + 15) / 16;  // m-tiles over output basis dim e
  constexpr int KC1 = (D + 3) / 4;    // K=4 chunks over input basis dim k

  __shared__ float lds_all[WAVES_PER_BLOCK][16 * LDS_PAD];
  const int lane = threadIdx.x & 31;
  const int wid  = threadIdx.x >> 5;
  const int col  = lane & 15;   // N index (channel for GEMM1 B, f for output)
  const int hi   = lane >> 4;   // lane-half: selects K pair / M-half per ISA layout
  float* lds = lds_all[wid];

  // ---------------- per-wave constant fragments (registers) ----------------
  // A-fragment of W^T: element (M=e, K=k) = W[k*D + e]; zero-padded via cndmask.
  v2f WT[MT][KC1];
#pragma unroll
  for (int mt = 0; mt < MT; ++mt)
#pragma unroll
    for (int kc = 0; kc < KC1; ++kc) {
      const int e  = mt * 16 + col;
      const int k0 = kc * 4 + 2 * hi;
      const bool v0 = (e < D) & (k0 < D);
      const bool v1 = (e < D) & (k0 + 1 < D);
      const int ec  = v0 | v1 ? (e < D ? e : 0) : 0;
      const int kc0 = k0 < D ? k0 : 0;
      const int kc1 = k0 + 1 < D ? k0 + 1 : 0;
      float a0 = W[kc0 * D + ec];
      float a1 = W[kc1 * D + ec];
      a0 = v0 ? a0 : 0.0f;
      a1 = v1 ? a1 : 0.0f;
      WT[mt][kc] = (v2f){a0, a1};
    }

  // B-fragments of cw: element (K=c, N=f) = cw[c*16 + f]
  v2f CW[4];
#pragma unroll
  for (int kc = 0; kc < 4; ++kc) {
    const int c0 = kc * 4 + 2 * hi;
    CW[kc] = (v2f){cw[c0 * 16 + col], cw[(c0 + 1) * 16 + col]};
  }

  // s[f] = sum_c cw[c,f]  (bias fold:  (Z + b)cw = Z cw + b (x) s)
  float s = 0.0f;
#pragma unroll
  for (int c = 0; c < 16; ++c) s += cw[c * 16 + col];

  // GEMM2 initial accumulator = b[e] * s[f] in C/D layout (M=r+8*hi, N=col)
  v8f BS[MT];
#pragma unroll
  for (int mt = 0; mt < MT; ++mt)
#pragma unroll
    for (int r = 0; r < 8; ++r) {
      const int e  = mt * 16 + r + 8 * hi;
      const int ec = e < D ? e : 0;
      BS[mt][r] = (e < D) ? b[ec] * s : 0.0f;
    }

  const int wave   = blockIdx.x * WAVES_PER_BLOCK + wid;
  const int nwaves = gridDim.x * WAVES_PER_BLOCK;

  for (int atom = wave; atom < n_atoms; atom += nwaves) {
    const long long ybase = base + (long long)atom * (D * 16);

    // prefetch next atom's tile into cache while we compute this one
    if (atom + nwaves < n_atoms)
      __builtin_prefetch(&x[base + (long long)(atom + nwaves) * (D * 16) + lane * 4], 0, 0);

    // ---- load B-fragments of Y (K=k rows, N=c cols), coalesced b32 loads ----
    // Interior chunks (kc*4+3 < D): guard folds away (uniform, constant after
    // unroll) -> plain loads. Boundary chunk: clamp address (still in this
    // atom's tile) and zero with v_cndmask -- no exec-mask branches.
    v2f BY[KC1];
#pragma unroll
    for (int kc = 0; kc < KC1; ++kc) {
      const int k0 = kc * 4 + 2 * hi;
      if (kc * 4 + 3 < D) {
        BY[kc] = (v2f){x[ybase + k0 * 16 + col], x[ybase + (k0 + 1) * 16 + col]};
      } else {
        const int kk0 = k0 < D ? k0 : 0;
        const int kk1 = k0 + 1 < D ? k0 + 1 : 0;
        float b0 = x[ybase + kk0 * 16 + col];
        float b1 = x[ybase + kk1 * 16 + col];
        b0 = (k0 < D) ? b0 : 0.0f;
        b1 = (k0 + 1 < D) ? b1 : 0.0f;
        BY[kc] = (v2f){b0, b1};
      }
    }

#pragma unroll
    for (int mt = 0; mt < MT; ++mt) {
      // ---- GEMM1: Z(tile mt) = W^T * Y ----
      v8f z = {};
#pragma unroll
      for (int kc = 0; kc < KC1; ++kc)
        z = __builtin_amdgcn_wmma_f32_16x16x4_f32(
            false, WT[mt][kc], false, BY[kc], (short)0, z, false, false);

      // ---- D-layout -> A-layout bounce through LDS (in-order per wave) ----
#pragma unroll
      for (int r = 0; r < 8; ++r)
        lds[(r + 8 * hi) * LDS_PAD + col] = z[r];
      asm volatile("s_wait_dscnt 0" ::: "memory");

      // ---- GEMM2: out(tile mt) = Z * cw + b (x) s ----
      v8f o = BS[mt];
#pragma unroll
      for (int kc = 0; kc < 4; ++kc) {
        const int c0 = kc * 4 + 2 * hi;
        // A-fragment: (M=col, K=c0..c0+1) -> 8B-aligned ds_load_b64
        const v2f a = *(const v2f*)&lds[col * LDS_PAD + c0];
        o = __builtin_amdgcn_wmma_f32_16x16x4_f32(
            false, a, false, CW[kc], (short)0, o, false, false);
      }

      // ---- store valid rows (disjoint coverage; every out row written once).
      // Interior rows: guard folds to constant-true -> unconditional store.
#pragma unroll
      for (int r = 0; r < 8; ++r) {
        const int e = mt * 16 + r + 8 * hi;
        if (mt * 16 + r + 8 < D) {
          out[ybase + e * 16 + col] = o[r];
        } else if (e < D) {
          out[ybase + e * 16 + col] = o[r];
        }
      }
    }
  }
}

// atom counts / dims from the reference
#define N_H 50000
#define N_C 30000
#define N_O 20000
#define D_H 14
#define D_C 30
#define D_O 30

extern "C" void kernel_launch(void* const* d_in, const int* in_sizes, int n_in,
                              void* d_out, int out_size, void* d_ws, size_t ws_size,
                              hipStream_t stream) {
  (void)in_sizes; (void)n_in; (void)out_size; (void)d_ws; (void)ws_size;
  const float* x    = (const float*)d_in[0];
  const float* W_h  = (const float*)d_in[1];
  const float* b_h  = (const float*)d_in[2];
  const float* cw_h = (const float*)d_in[3];
  const float* W_c  = (const float*)d_in[4];
  const float* b_c  = (const float*)d_in[5];
  const float* cw_c = (const float*)d_in[6];
  const float* W_o  = (const float*)d_in[7];
  const float* b_o  = (const float*)d_in[8];
  const float* cw_o = (const float*)d_in[9];
  float* out = (float*)d_out;

  const long long base_h = 0;
  const long long base_c = (long long)N_H * D_H * 16;            // 11,200,000
  const long long base_o = base_c + (long long)N_C * D_C * 16;   // 25,600,000

  // ~4 atoms per wave so per-wave constant-fragment setup amortizes
  auto blocks = [](int n) { return (n + WAVES_PER_BLOCK * 4 - 1) / (WAVES_PER_BLOCK * 4); };

  node_linear_wmma<D_H><<<blocks(N_H), 256, 0, stream>>>(x, W_h, b_h, cw_h, out, N_H, base_h);
  node_linear_wmma<D_C><<<blocks(N_C), 256, 0, stream>>>(x, W_c, b_c, cw_c, out, N_C, base_c);
  node_linear_wmma<D_O><<<blocks(N_O), 256, 0, stream>>>(x, W_o, b_o, cw_o, out, N_O, base_o);
}